// FourierShellCorrelation_81140522156479
// MI455X (gfx1250) — compile-verified
//
#include <hip/hip_runtime.h>

// Fourier Shell Correlation on MI455X (gfx1250).
//
// 2D FFT as four-step 512 = 16 x 32, both DFT stages on V_WMMA_F32_16X16X4_F32
// (exact fp32 matrix pipes). One wave32 = one 512-point complex DFT:
//   stage1: Y(16x32) = X(16x32) * F32(32x32)   -> 64 wmma
//   twiddle W_512^{n1*k2} from an LDS table, pair-packed LDS bounce
//   stage2: Out(16x32) = F16(16x16) * Z(16x32) -> 32 wmma
// All LDS matrix planes are stored PAIR-PACKED (float2 = rows {2p, 2p+1} at
// column n) so each WMMA B-fragment is exactly one ds_load_b64 into the
// even-aligned VGPR pair the instruction requires -- no repacking moves.
//
// Pass1: row FFTs of t and (z - t), stored transposed into d_ws (64 MB).
// Pass2: column FFTs fused with |.|^2 radial binning (ds_add_f32 bins, one
//        global atomic flush per workgroup).
// Pass3: nre = es/max(ns,1e-8), sum shells, mean over batch -> d_out[0].
//
// d_ws requirement: 2*16*512*512 float2 (64 MB) + 2*16*512 floats (64 KB).

typedef __attribute__((ext_vector_type(2))) float v2f;
typedef __attribute__((ext_vector_type(8))) float v8f;

namespace {
constexpr int kB = 16;
constexpr int kH = 512;
constexpr int kW = 512;
constexpr int kBins = 511;
constexpr float kPi = 3.14159265358979323846f;
constexpr int kWaves = 4;            // waves per block (one FFT line per wave)
constexpr int kBlock = kWaves * 32;  // wave32
}  // namespace

static __device__ __forceinline__ v8f wmma4(v2f a, v2f b, v8f c) {
  // D = A(16x4 f32) * B(4x16 f32) + C(16x16 f32); f32 WMMA has no A/B NEG.
  return __builtin_amdgcn_wmma_f32_16x16x4_f32(false, a, false, b, (short)0, c,
                                               false, false);
}

static __device__ __forceinline__ v2f ldpair(const float2* p) {
  float2 f = *p;
  v2f r;
  r.x = f.x;
  r.y = f.y;
  return r;
}

// Pair-packed DFT-32 planes (W = c + i*s, s = -sin) and the W_512 twiddle
// grid. pc/ps/pn[p*32+n] = {plane(row 2p, n), plane(row 2p+1, n)};
// tw[m*32+n] = {cos(2pi*m*n/512), sin(2pi*m*n/512)}.
static __device__ __forceinline__ void build_tables(float2* pc, float2* ps,
                                                    float2* pn, float2* tw) {
  for (int i = threadIdx.x; i < 512; i += kBlock) {
    int p = i >> 5, n = i & 31;
    float a0 = (2.0f * kPi / 32.0f) * (float)((2 * p) * n);
    float a1 = (2.0f * kPi / 32.0f) * (float)((2 * p + 1) * n);
    float s0, c0, s1, c1;
    __sincosf(a0, &s0, &c0);
    __sincosf(a1, &s1, &c1);
    pc[i] = make_float2(c0, c1);
    ps[i] = make_float2(-s0, -s1);
    pn[i] = make_float2(s0, s1);
    float at = (2.0f * kPi / 512.0f) * (float)(p * n);  // p == m here
    float st, ct;
    __sincosf(at, &st, &ct);
    tw[i] = make_float2(ct, st);
  }
}

// One wave: full 512-point complex DFT of Xmat[n1][n2] = x[n1 + 16*n2].
// Output O = {Or(tile0), Oi(tile0), Or(tile1), Oi(tile1)} in WMMA C-layout:
// element (m = r + 8*(lane>=16), n = (lane&15) + 16*tile) = X[n + 32*m].
// zrp/zip: per-wave pair-packed Z staging planes, 8*32 float2 each.
// Contains one __syncthreads (uniform across the block).
static __device__ __forceinline__ void fft512_core(
    const v2f Xr[8], const v2f Xi[8], const float2* __restrict__ pc,
    const float2* __restrict__ ps, const float2* __restrict__ pn,
    const float2* __restrict__ tw, float2* __restrict__ zrp,
    float2* __restrict__ zip, int laneM, int laneHi, v8f O[4]) {
  // Stage-2 A fragments: F16[m=k1][k=n1] = W_16^{k1*n1}; A layout: lanes 0-15
  // hold K pair {4j,4j+1}, lanes 16-31 hold {4j+2,4j+3}, M = lane&15.
  v2f Ac[4], As[4], Ans[4];
#pragma unroll
  for (int j = 0; j < 4; ++j) {
    int k0 = 4 * j + 2 * laneHi;
    float a0 = (2.0f * kPi / 16.0f) * (float)(laneM * k0);
    float a1 = (2.0f * kPi / 16.0f) * (float)(laneM * (k0 + 1));
    float s0, c0, s1, c1;
    __sincosf(a0, &s0, &c0);
    __sincosf(a1, &s1, &c1);
    Ac[j].x = c0;  Ac[j].y = c1;
    As[j].x = -s0; As[j].y = -s1;
    Ans[j].x = s0; Ans[j].y = s1;
  }

  // Stage 1: Y(16x32) = X(16x32) * F32(32x32), complex.
  v8f Yr0 = {}, Yr1 = {}, Yi0 = {}, Yi1 = {};
#pragma unroll
  for (int j = 0; j < 8; ++j) {
    int p = 2 * j + laneHi;  // pair index: rows {4j+2*laneHi, +1}
#pragma unroll
    for (int tc = 0; tc < 2; ++tc) {
      int n = laneM + 16 * tc;
      v2f bc = ldpair(&pc[p * 32 + n]);
      v2f bs = ldpair(&ps[p * 32 + n]);
      v2f bn = ldpair(&pn[p * 32 + n]);
      v8f& yr = tc ? Yr1 : Yr0;
      v8f& yi = tc ? Yi1 : Yi0;
      yr = wmma4(Xr[j], bc, yr);  // Yr += Xr*c
      yr = wmma4(Xi[j], bn, yr);  // Yr += Xi*(-s)
      yi = wmma4(Xr[j], bs, yi);  // Yi += Xr*s
      yi = wmma4(Xi[j], bc, yi);  // Yi += Xi*c
    }
  }

  // Twiddle by W_512^{m*n} (table), store pair-packed planes for stage 2.
#pragma unroll
  for (int tc = 0; tc < 2; ++tc) {
    v8f yr = tc ? Yr1 : Yr0;
    v8f yi = tc ? Yi1 : Yi0;
    int n = laneM + 16 * tc;
    float zr[8], zi[8];
#pragma unroll
    for (int r = 0; r < 8; ++r) {
      int m = r + 8 * laneHi;
      float2 w = tw[m * 32 + n];  // (yr + i*yi) * (w.x - i*w.y)
      zr[r] = yr[r] * w.x + yi[r] * w.y;
      zi[r] = yi[r] * w.x - yr[r] * w.y;
    }
#pragma unroll
    for (int rp = 0; rp < 4; ++rp) {  // rows (m, m+1), m = 2*rp + 8*laneHi
      int pair = rp + 4 * laneHi;
      zrp[pair * 32 + n] = make_float2(zr[2 * rp], zr[2 * rp + 1]);
      zip[pair * 32 + n] = make_float2(zi[2 * rp], zi[2 * rp + 1]);
    }
  }
  __syncthreads();

  // Stage 2: Out(16x32) = F16(16x16) * Z(16x32), complex.
  v8f Or0 = {}, Or1 = {}, Oi0 = {}, Oi1 = {};
#pragma unroll
  for (int j = 0; j < 4; ++j) {
    int p = 2 * j + laneHi;  // pair index: K rows {4j+2*laneHi, +1}
#pragma unroll
    for (int tc = 0; tc < 2; ++tc) {
      int n = laneM + 16 * tc;
      v2f Zr = ldpair(&zrp[p * 32 + n]);
      v2f Zi = ldpair(&zip[p * 32 + n]);
      v8f& orr = tc ? Or1 : Or0;
      v8f& oii = tc ? Oi1 : Oi0;
      orr = wmma4(Ac[j], Zr, orr);   // Or += c*Zr
      orr = wmma4(Ans[j], Zi, orr);  // Or += (-s)*Zi
      oii = wmma4(As[j], Zr, oii);   // Oi += s*Zr
      oii = wmma4(Ac[j], Zi, oii);   // Oi += c*Zi
    }
  }
  O[0] = Or0; O[1] = Oi0; O[2] = Or1; O[3] = Oi1;
}

// Pass 1: row FFTs of t (sel=0) and z - t (sel=1); transposed store so pass 2
// reads contiguous lines. interm[(sel*B+b)][k_w][row].
__global__ __launch_bounds__(kBlock) void fsc_fft_rows(
    const float2* __restrict__ zin, const float2* __restrict__ tin,
    float2* __restrict__ interm) {
  __shared__ float2 pc[512], ps[512], pn[512], tw[512];
  __shared__ float2 zrp[kWaves][8 * 32], zip[kWaves][8 * 32];
  const int wv = threadIdx.x >> 5;
  const int lane = threadIdx.x & 31;
  const int laneM = lane & 15, laneHi = lane >> 4;
  build_tables(pc, ps, pn, tw);

  const int row = blockIdx.x * kWaves + wv;
  const int b = blockIdx.y;
  const int sel = blockIdx.z;

  const float2* trow = tin + ((size_t)(b * kH + row)) * kW;
  const float2* zrow = zin + ((size_t)(b * kH + row)) * kW;
  __builtin_prefetch(trow + lane * 16, 0, 0);  // global_prefetch_b8
  if (sel) __builtin_prefetch(zrow + lane * 16, 0, 0);
  __syncthreads();

  v2f Xr[8], Xi[8];
#pragma unroll
  for (int j = 0; j < 8; ++j) {
    int n2 = 4 * j + 2 * laneHi;
    int i0 = laneM + 16 * n2;  // Xmat[n1][n2] = x[n1 + 16*n2]
    float2 v0 = trow[i0], v1 = trow[i0 + 16];
    if (sel) {
      float2 a0 = zrow[i0], a1 = zrow[i0 + 16];
      v0.x = a0.x - v0.x; v0.y = a0.y - v0.y;
      v1.x = a1.x - v1.x; v1.y = a1.y - v1.y;
    }
    Xr[j].x = v0.x; Xr[j].y = v1.x;
    Xi[j].x = v0.y; Xi[j].y = v1.y;
  }

  v8f O[4];
  fft512_core(Xr, Xi, pc, ps, pn, tw, zrp[wv], zip[wv], laneM, laneHi, O);

  float2* obase = interm + ((size_t)(sel * kB + b)) * kH * kW;
#pragma unroll
  for (int tc = 0; tc < 2; ++tc) {
    v8f orr = O[2 * tc], oii = O[2 * tc + 1];
    int k2 = laneM + 16 * tc;
#pragma unroll
    for (int r = 0; r < 8; ++r) {
      int k1 = r + 8 * laneHi;
      int k = k2 + 32 * k1;  // frequency along W
      obase[(size_t)k * kH + row] = make_float2(orr[r], oii[r]);
    }
  }
}

// Pass 2: column FFTs + fused |.|^2 radial binning.
// sel=0 accumulates ns (|FFT(t)|^2), sel=1 accumulates es (|FFT(z-t)|^2).
__global__ __launch_bounds__(kBlock) void fsc_fft_cols(
    const float2* __restrict__ interm, float* __restrict__ acc_ns,
    float* __restrict__ acc_es) {
  __shared__ float2 pc[512], ps[512], pn[512], tw[512];
  __shared__ float2 zrp[kWaves][8 * 32], zip[kWaves][8 * 32];
  __shared__ float binacc[512];
  const int wv = threadIdx.x >> 5;
  const int lane = threadIdx.x & 31;
  const int laneM = lane & 15, laneHi = lane >> 4;
  build_tables(pc, ps, pn, tw);
  for (int i = threadIdx.x; i < 512; i += kBlock) binacc[i] = 0.0f;

  const int v = blockIdx.x * kWaves + wv;  // frequency along W (the line)
  const int b = blockIdx.y;
  const int sel = blockIdx.z;

  const float2* linep = interm + (((size_t)(sel * kB + b)) * kW + v) * kH;
  __builtin_prefetch(linep + lane * 16, 0, 0);  // global_prefetch_b8
  __syncthreads();

  v2f Xr[8], Xi[8];
#pragma unroll
  for (int j = 0; j < 8; ++j) {
    int n2 = 4 * j + 2 * laneHi;
    int i0 = laneM + 16 * n2;
    float2 v0 = linep[i0], v1 = linep[i0 + 16];
    Xr[j].x = v0.x; Xr[j].y = v1.x;
    Xi[j].x = v0.y; Xi[j].y = v1.y;
  }

  v8f O[4];
  fft512_core(Xr, Xi, pc, ps, pn, tw, zrp[wv], zip[wv], laneM, laneHi, O);

  // shell id: ellipse = ((u-256)/256)^2 + ((v-256)/256)^2; searchsorted-right
  // against ((k+1)/511)^2 == floor(511*sqrt(e)); bins >= 511 dropped.
  const float fy = (float)(v - 256) * (1.0f / 256.0f);
#pragma unroll
  for (int tc = 0; tc < 2; ++tc) {
    v8f orr = O[2 * tc], oii = O[2 * tc + 1];
    int k2 = laneM + 16 * tc;
#pragma unroll
    for (int r = 0; r < 8; ++r) {
      int k1 = r + 8 * laneHi;
      int u = k2 + 32 * k1;  // frequency along H
      float fx = (float)(u - 256) * (1.0f / 256.0f);
      float e = fx * fx + fy * fy;
      int bin = (int)floorf(511.0f * sqrtf(e));
      float mag = orr[r] * orr[r] + oii[r] * oii[r];
      if (bin < kBins) atomicAdd(&binacc[bin], mag);  // ds_add_f32
    }
  }
  __syncthreads();
  float* acc = sel ? acc_es : acc_ns;
  for (int i = threadIdx.x; i < 512; i += kBlock) {
    float s = binacc[i];
    if (s != 0.0f) atomicAdd(&acc[b * 512 + i], s);  // global_atomic_add_f32
  }
}

__global__ void fsc_zero(float* __restrict__ acc) {
  int i = blockIdx.x * 256 + threadIdx.x;
  if (i < 2 * kB * 512) acc[i] = 0.0f;
}

__global__ __launch_bounds__(256) void fsc_reduce(
    const float* __restrict__ acc_ns, const float* __restrict__ acc_es,
    float* __restrict__ out) {
  __shared__ float red[256];
  float p = 0.0f;
  for (int i = threadIdx.x; i < kB * kBins; i += 256) {
    int b = i / kBins, s = i % kBins;
    float nsv = acc_ns[b * 512 + s];
    float esv = acc_es[b * 512 + s];
    p += esv / fmaxf(nsv, 1e-8f);
  }
  red[threadIdx.x] = p;
  __syncthreads();
  for (int off = 128; off > 0; off >>= 1) {
    if ((int)threadIdx.x < off) red[threadIdx.x] += red[threadIdx.x + off];
    __syncthreads();
  }
  if (threadIdx.x == 0) out[0] = red[0] / (float)kB;
}

extern "C" void kernel_launch(void* const* d_in, const int* in_sizes, int n_in,
                              void* d_out, int out_size, void* d_ws,
                              size_t ws_size, hipStream_t stream) {
  (void)in_sizes; (void)n_in; (void)out_size; (void)ws_size;
  const float2* z = (const float2*)d_in[0];  // z_  [B,H,W,2] f32
  const float2* t = (const float2*)d_in[1];  // t_  [B,H,W,2] f32

  float2* interm = (float2*)d_ws;  // [2][B][kW][kH] transposed spectra (64 MB)
  const size_t interm_elems = (size_t)2 * kB * kH * kW;
  float* acc = (float*)(interm + interm_elems);  // ns[B*512] then es[B*512]
  float* acc_ns = acc;
  float* acc_es = acc + kB * 512;

  fsc_zero<<<(2 * kB * 512 + 255) / 256, 256, 0, stream>>>(acc);
  fsc_fft_rows<<<dim3(kH / kWaves, kB, 2), kBlock, 0, stream>>>(z, t, interm);
  fsc_fft_cols<<<dim3(kW / kWaves, kB, 2), kBlock, 0, stream>>>(interm, acc_ns,
                                                                acc_es);
  fsc_reduce<<<1, 256, 0, stream>>>(acc_ns, acc_es, (float*)d_out);
}